// SpatialNumDualDescriptorAB4_36790689858203
// MI455X (gfx1250) — compile-verified
//
#include <hip/hip_runtime.h>

typedef float v2f __attribute__((ext_vector_type(2)));
typedef float v4f __attribute__((ext_vector_type(4)));
typedef float v8f __attribute__((ext_vector_type(8)));

#define M_DIM 32
#define NW    15      // windows per dim (16 - 2 + 1)
#define LT    81      // 3^4 basis functions
#define BSZ   4

// Strides of arr_batch [4][16][16][16][16][32] in elements
#define STR_S  32
#define STR_R  512
#define STR_Q  8192
#define STR_P  131072
#define STR_B  2097152

// ---------------------------------------------------------------------------
// Kernel 0: Psi[j][m] = sum_n M[n][m] * Bbasis[j][n]; zero the S accumulator.
// ---------------------------------------------------------------------------
__global__ void psi_init_kernel(const float* __restrict__ Mmat,
                                const float* __restrict__ Bbasis,
                                float* __restrict__ Psi,
                                float* __restrict__ S) {
  int t = threadIdx.x;
  for (int e = t; e < LT * M_DIM; e += blockDim.x) {
    int j = e >> 5, m = e & 31;
    float acc = 0.f;
#pragma unroll
    for (int n = 0; n < 32; ++n)
      acc += Mmat[n * 32 + m] * Bbasis[j * 32 + n];
    Psi[e] = acc;
  }
  for (int e = t; e < BSZ * LT; e += blockDim.x) S[e] = 0.f;
}

// ---------------------------------------------------------------------------
// Kernel 1: one block per (b, p', q', r').
//   stage 1: colsum[s][m] = sum over 8 (dp,dq,dr) combos (b128 loads)
//            psiB[col][k] = Psi[jbase+col][k] for col<3, else 0
//   stage 2: pooled[row][m] = colsum[row][m] + colsum[row+1][m]  (row<15)
//   stage 3: wave 0: 8 chained V_WMMA_F32_16X16X4_F32 (all operands from LDS),
//            static-mask diagonal select, per-lane pre-reduction,
//            <=6 atomics into S.
// ---------------------------------------------------------------------------
__global__ void window_wmma_kernel(const float* __restrict__ arr,
                                   const float* __restrict__ Psi,
                                   float* __restrict__ S) {
  __shared__ float colsum[16][M_DIM];
  __shared__ float pooled[16][M_DIM];
  __shared__ float psiB[16][M_DIM];

  const int bid = blockIdx.x;
  const int r3 = bid % NW;
  const int q2 = (bid / NW) % NW;
  const int p1 = (bid / (NW * NW)) % NW;
  const int b  = bid / (NW * NW * NW);

  const int jbase = (((p1 % 3) * 3 + (q2 % 3)) * 3 + (r3 % 3)) * 3;

  const float* tile = arr + (size_t)b * STR_B + (size_t)p1 * STR_P +
                      (size_t)q2 * STR_Q + (size_t)r3 * STR_R;

  const int t = threadIdx.x;        // 256 threads

  // --- stage 1a: column sums over the 8 (dp,dq,dr) corner combos ---------
  if (t < 128) {
    const int m4 = (t & 7) * 4;     // m in steps of 4
    const int s  = t >> 3;          // 0..15
    const float* p0 = tile + s * STR_S + m4;
    v4f acc = {0.f, 0.f, 0.f, 0.f};
#pragma unroll
    for (int c = 0; c < 8; ++c) {
      const int dp = (c >> 2) & 1, dq = (c >> 1) & 1, dr = c & 1;
      acc += *(const v4f*)(p0 + dp * STR_P + dq * STR_Q + dr * STR_R);
    }
    *(v4f*)&colsum[s][m4] = acc;
  }

  // --- stage 1b: stage B tile (3 live Psi rows, rest zero) in LDS --------
#pragma unroll
  for (int half = 0; half < 2; ++half) {
    const int e   = t + 256 * half;       // 0..511
    const int col = e >> 5, k = e & 31;
    psiB[col][k] = (col < 3) ? Psi[(jbase + col) * M_DIM + k] : 0.f;
  }
  __syncthreads();

  // --- stage 2: 2x2x2x2 pooled sums (row 15 = zero pad) -------------------
#pragma unroll
  for (int half = 0; half < 2; ++half) {
    const int e   = t + 256 * half;       // 0..511
    const int row = e >> 5, m = e & 31;
    pooled[row][m] = (row < 15) ? (colsum[row][m] + colsum[row + 1][m]) : 0.f;
  }
  __syncthreads();

  // --- stage 3: wave 0 does the WMMA chain (EXEC all ones) ----------------
  if (t < 32) {
    const int lane = t;
    const int hi   = lane >> 4;     // 0: K {0,1}/rows 0-7   1: K {2,3}/rows 8-15
    const int col  = lane & 15;

    v8f acc{};
#pragma unroll
    for (int k0 = 0; k0 < 32; k0 += 4) {
      const int ka = k0 + 2 * hi;
      const v2f a  = *(const v2f*)&pooled[col][ka];
      const v2f bb = *(const v2f*)&psiB[col][ka];
      acc = __builtin_amdgcn_wmma_f32_16x16x4_f32(
          /*neg_a=*/false, a, /*neg_b=*/false, bb,
          /*c_mod=*/(short)0, acc, /*reuse_a=*/false, /*reuse_b=*/false);
    }

    if (col < 3) {
      // Static per-(hi,col) row-selection masks: bit r set iff
      // (r + 8*hi) % 3 == col and (r + 8*hi) < 15.
      //   hi=0: col0 -> 0x49 {0,3,6}, col1 -> 0x92 {1,4,7}, col2 -> 0x24 {2,5}
      //   hi=1: col0 -> 0x12 {1,4},   col1 -> 0x24 {2,5},   col2 -> 0x49 {0,3,6}
      const unsigned mask =
          (unsigned)((0x0000492412249249ULL >> ((hi * 3 + col) * 8)) & 0xFFu);
      float ssum = 0.f;
#pragma unroll
      for (int r = 0; r < 8; ++r)
        ssum += (mask & (1u << r)) ? acc[r] : 0.f;
      atomicAdd(&S[b * LT + jbase + col], ssum);
    }
  }
}

// ---------------------------------------------------------------------------
// Kernel 2: out[b][n] = (1/(16*50625)) * sum_j Acoeff[n][j] * S[b][j]
// ---------------------------------------------------------------------------
__global__ void out_kernel(const float* __restrict__ Acoeff,
                           const float* __restrict__ S,
                           float* __restrict__ out) {
  const int t = threadIdx.x;
  if (t >= BSZ * M_DIM) return;
  const int b = t >> 5, n = t & 31;
  float acc = 0.f;
#pragma unroll
  for (int j = 0; j < LT; ++j)
    acc += Acoeff[n * LT + j] * S[b * LT + j];
  out[b * M_DIM + n] = acc * (1.0f / (16.0f * 50625.0f));
}

extern "C" void kernel_launch(void* const* d_in, const int* in_sizes, int n_in,
                              void* d_out, int out_size, void* d_ws, size_t ws_size,
                              hipStream_t stream) {
  (void)in_sizes; (void)n_in; (void)out_size; (void)ws_size;
  const float* arr    = (const float*)d_in[0];   // [4,16,16,16,16,32]
  const float* Mmat   = (const float*)d_in[1];   // [32,32]
  const float* Acoeff = (const float*)d_in[2];   // [32,81]
  const float* Bbasis = (const float*)d_in[3];   // [81,32]
  float* out = (float*)d_out;                    // [4,32]

  float* Psi = (float*)d_ws;                     // 81*32 floats
  float* S   = Psi + LT * M_DIM;                 // 4*81 floats

  psi_init_kernel<<<1, 256, 0, stream>>>(Mmat, Bbasis, Psi, S);
  window_wmma_kernel<<<BSZ * NW * NW * NW, 256, 0, stream>>>(arr, Psi, S);
  out_kernel<<<1, 128, 0, stream>>>(Acoeff, S, out);
}